// TechnicalIndicators_1529008358057
// MI455X (gfx1250) — compile-verified
//
#include <hip/hip_runtime.h>

typedef float f4  __attribute__((ext_vector_type(4)));
typedef float v2f __attribute__((ext_vector_type(2)));
typedef float v8f __attribute__((ext_vector_type(8)));

#define B_DIM   64
#define S_DIM   4096
#define C_DIM   64
#define NCHUNK  8
#define SCHUNK  (S_DIM / NCHUNK)   /* 512 rows per chunk */
#define OUT_C   320
#define OUT_F4  (OUT_C / 4)        /* 80 */
#define IN_F4   (C_DIM / 4)        /* 16 */

// stat indices: 0=sum 1=sumsq 2=gain 3=loss 4=last12 5=last26

__device__ __forceinline__ f4 relu4(f4 v) {
  f4 r;
  r.x = fmaxf(v.x, 0.0f); r.y = fmaxf(v.y, 0.0f);
  r.z = fmaxf(v.z, 0.0f); r.w = fmaxf(v.w, 0.0f);
  return r;
}

// Kernel 1: stream input chunk, copy prices to out[:, :, 0:64], accumulate 6
// per-channel partial stats, WMMA-reduce the 16 in-block partials, spill to ws.
__global__ __launch_bounds__(256) void ti_stats_kernel(const float* __restrict__ price,
                                                       float* __restrict__ out,
                                                       float* __restrict__ ws) {
  const int blk   = blockIdx.x;
  const int b     = blk >> 3;
  const int chunk = blk & 7;
  const int cs    = chunk * SCHUNK;
  const int t     = threadIdx.x;
  const int j     = t & 15;          // float4 column (channels 4j..4j+3)
  const int sg    = t >> 4;          // 0..15 row-group
  const int rows  = SCHUNK / 16;     // 32 contiguous rows per thread

  const f4* in4  = (const f4*)price;
  f4*       out4 = (f4*)out;

  f4 sum = {0,0,0,0}, sumsq = {0,0,0,0}, gain = {0,0,0,0}, loss = {0,0,0,0};
  f4 s12 = {0,0,0,0}, s26 = {0,0,0,0};

  const int    s0   = cs + sg * rows;
  const size_t base = (size_t)b * S_DIM;

  f4 prev = {0,0,0,0};
  if (s0 > 0) prev = in4[(base + (size_t)(s0 - 1)) * IN_F4 + j];

  for (int i = 0; i < rows; ++i) {
    const int s = s0 + i;
    f4 x = __builtin_nontemporal_load(&in4[(base + (size_t)s) * IN_F4 + j]);
    sum   += x;
    sumsq += x * x;
    if (s > 0) {
      f4 d = x - prev;
      gain += relu4(d);
      loss += relu4(-d);
    }
    if (s >= S_DIM - 12) s12 += x;
    if (s >= S_DIM - 26) s26 += x;
    // raw price copy -> output channels [0, 64)
    __builtin_nontemporal_store(x, &out4[(base + (size_t)s) * OUT_F4 + j]);
    prev = x;
  }

  __shared__ __align__(16) float lds[6][16][64];
  *(f4*)&lds[0][sg][j * 4] = sum;
  *(f4*)&lds[1][sg][j * 4] = sumsq;
  *(f4*)&lds[2][sg][j * 4] = gain;
  *(f4*)&lds[3][sg][j * 4] = loss;
  *(f4*)&lds[4][sg][j * 4] = s12;
  *(f4*)&lds[5][sg][j * 4] = s26;
  __syncthreads();

  // WMMA column-sum reduction: D = Ones(16x4) x P(4x16) + C, chained over 4
  // K-slices to fold 16 row-group partials. All lanes of acc[0] hold
  // colsum[lane & 15] (rows of D identical), so extraction is layout-robust.
  const int wave = t >> 5;
  const int lane = t & 31;
  const int half = lane >> 4;
  const int cl   = lane & 15;
  const v2f ones = {1.0f, 1.0f};

  for (int j3 = 0; j3 < 3; ++j3) {
    const int job  = wave * 3 + j3;      // 24 jobs = 6 stats x 4 channel-groups
    const int stat = job % 6;
    const int cg   = job / 6;
    const int c    = cg * 16 + cl;
    v8f acc = {0,0,0,0,0,0,0,0};
#pragma unroll
    for (int kk = 0; kk < 4; ++kk) {
      const int k0 = kk * 4 + half * 2;  // bijective K assignment; sum is commutative
      v2f bm;
      bm.x = lds[stat][k0][c];
      bm.y = lds[stat][k0 + 1][c];
      acc = __builtin_amdgcn_wmma_f32_16x16x4_f32(false, ones, false, bm,
                                                  (short)0, acc, false, false);
    }
    if (lane < 16) {
      ws[(((size_t)b * NCHUNK + chunk) * 6 + stat) * 64 + c] = acc[0];
    }
  }
}

// Kernel 2: WMMA-reduce the 8 chunk partials, compute indicators, stream the
// 256 broadcast channels for this block's S-chunk with B128 NT stores.
__global__ __launch_bounds__(256) void ti_write_kernel(const float* __restrict__ ws,
                                                       float* __restrict__ out) {
  const int blk   = blockIdx.x;
  const int b     = blk >> 3;
  const int chunk = blk & 7;
  const int cs    = chunk * SCHUNK;
  const int t     = threadIdx.x;

  __shared__ float red[6][64];
  __shared__ __align__(16) float bcast[256];

  const int wave = t >> 5;
  const int lane = t & 31;
  const int half = lane >> 4;
  const int cl   = lane & 15;
  const v2f ones = {1.0f, 1.0f};

  for (int j3 = 0; j3 < 3; ++j3) {
    const int job  = wave * 3 + j3;
    const int stat = job % 6;
    const int cg   = job / 6;
    const int c    = cg * 16 + cl;
    v8f acc = {0,0,0,0,0,0,0,0};
#pragma unroll
    for (int kk = 0; kk < 2; ++kk) {
      const int k0 = kk * 4 + half * 2;  // k in [0,8) bijective
      v2f bm;
      bm.x = ws[(((size_t)b * NCHUNK + k0)     * 6 + stat) * 64 + c];
      bm.y = ws[(((size_t)b * NCHUNK + k0 + 1) * 6 + stat) * 64 + c];
      acc = __builtin_amdgcn_wmma_f32_16x16x4_f32(false, ones, false, bm,
                                                  (short)0, acc, false, false);
    }
    if (lane < 16) red[stat][c] = acc[0];
  }
  __syncthreads();

  if (t < 64) {
    const float sum   = red[0][t];
    const float sumsq = red[1][t];
    const float gain  = red[2][t];
    const float loss  = red[3][t];
    const float s12   = red[4][t];
    const float s26   = red[5][t];

    const float invS = 1.0f / (float)S_DIM;
    const float mean = sum * invS;
    const float var  = fmaxf(sumsq * invS - mean * mean, 0.0f);
    const float sd   = sqrtf(var);

    const float invD     = 1.0f / (float)(S_DIM - 1);
    const float avg_gain = gain * invD;
    const float avg_loss = loss * invD;
    const float rs       = avg_gain / (avg_loss + 1e-7f);
    const float rsi      = 100.0f - 100.0f / (1.0f + rs);

    const float macd = s12 * (1.0f / 12.0f) - s26 * (1.0f / 26.0f);

    bcast[t]       = rsi;                 // channels  64..127
    bcast[64 + t]  = macd;                // channels 128..191
    bcast[128 + t] = mean + 2.0f * sd;    // channels 192..255
    bcast[192 + t] = mean - 2.0f * sd;    // channels 256..319
  }
  __syncthreads();

  const int jj = t & 63;   // float4 column within the 256 broadcast channels
  const int sr = t >> 6;   // 4 s-rows in flight
  const f4 v = ((const f4*)bcast)[jj];
  f4* out4 = (f4*)out;
  const size_t base = (size_t)b * S_DIM;

  for (int s = cs + sr; s < cs + SCHUNK; s += 4) {
    __builtin_nontemporal_store(v, &out4[(base + (size_t)s) * OUT_F4 + 16 + jj]);
  }
}

extern "C" void kernel_launch(void* const* d_in, const int* in_sizes, int n_in,
                              void* d_out, int out_size, void* d_ws, size_t ws_size,
                              hipStream_t stream) {
  (void)in_sizes; (void)n_in; (void)out_size; (void)ws_size;
  const float* price = (const float*)d_in[0];
  float* out = (float*)d_out;
  float* ws  = (float*)d_ws;   // needs 64*8*6*64*4 = 786432 bytes

  ti_stats_kernel<<<dim3(B_DIM * NCHUNK), dim3(256), 0, stream>>>(price, out, ws);
  ti_write_kernel<<<dim3(B_DIM * NCHUNK), dim3(256), 0, stream>>>(ws, out);
}